// ModifiedDGCNNExtractor_29970281791924
// MI455X (gfx1250) — compile-verified
//
#include <hip/hip_runtime.h>
#include <float.h>

// ---------------------------------------------------------------------------
// ModifiedDGCNNExtractor for MI455X (gfx1250, wave32).
// fp32 WMMA (v_wmma_f32_16x16x4_f32) for both the kNN inner-product GEMM and
// the edge-conv GEMM. Whole working set (<20 MB) lives in L2 (192 MB), so the
// kernels are compute/latency bound; fp32 matrix path keeps top-k ranking and
// layer-norm numerically faithful.
//
// v2 changes:
//  - edge-conv: per-lane running-max flush => ~4x fewer float atomicMax ops
//    (atomics were 16x the WMMA count in stage 4, the true hot spot).
//  - edge-conv: 4 independent waves per 128-thread workgroup (better
//    waves/WGP under the 32-workgroup cap, 4x fewer dispatch packets).
//  - kNN: global_prefetch of the next 16-column tile overlaps the top-k
//    insertion phase with the next tile's B-fragment fetch.
// ---------------------------------------------------------------------------

#define B_    16
#define N_    2048
#define KNN   20
#define EPS_  1e-5f
#define SLOPE 0.2f

typedef __attribute__((ext_vector_type(2))) float v2f;
typedef __attribute__((ext_vector_type(8))) float v8f;

// ---------------- transpose (B,3,N) -> (B,N,4) padded ----------------------
__global__ void k_transpose_pad(const float* __restrict__ x, float* __restrict__ xt) {
  int i = blockIdx.x * blockDim.x + threadIdx.x;
  if (i >= B_ * N_) return;
  int b = i / N_, n = i % N_;
  const float* xp = x + (size_t)b * 3 * N_ + n;
  float* o = xt + (size_t)i * 4;
  o[0] = xp[0]; o[1] = xp[N_]; o[2] = xp[2 * N_]; o[3] = 0.f;
}

// ---------------- squared norms --------------------------------------------
__global__ void k_norms(const float* __restrict__ xf, int CP, float* __restrict__ xx) {
  int i = blockIdx.x * blockDim.x + threadIdx.x;
  if (i >= B_ * N_) return;
  const float* p = xf + (size_t)i * CP;
  float s = 0.f;
  for (int c = 0; c < CP; ++c) s += p[c] * p[c];
  xx[i] = s;
}

// ---------------- fill ------------------------------------------------------
__global__ void k_fill(float* __restrict__ p, int count, float v) {
  int i = blockIdx.x * blockDim.x + threadIdx.x;
  if (i < count) p[i] = v;
}

// ---------------- kNN: WMMA inner products + register top-20 ---------------
// One wave handles 32 points (2 WMMA row tiles); streams 16-column tiles.
// Ranking key: 2*inner - xx[m]  (row-constant -xx[n] dropped, same ordering).
template <int CP>
__global__ __launch_bounds__(32) void k_knn(const float* __restrict__ xf,
                                            const float* __restrict__ xx,
                                            int* __restrict__ nbr) {
  constexpr int S = CP / 4;
  __shared__ float lds[32 * 16];
  int blk  = blockIdx.x;
  int b    = blk / (N_ / 32);
  int r0   = (blk % (N_ / 32)) * 32;
  int lane = threadIdx.x;
  int lo   = lane & 15, hi = lane >> 4;
  const float* xb = xf + (size_t)b * N_ * CP;

  // A fragments: rows r0..r0+31 (two 16-row tiles), K = feature channel.
  v2f A[2][S];
#pragma unroll
  for (int h = 0; h < 2; ++h) {
    const float* rp = xb + (size_t)(r0 + h * 16 + lo) * CP;
#pragma unroll
    for (int s = 0; s < S; ++s) {
      int c = 4 * s + 2 * hi;
      A[h][s].x = rp[c];
      A[h][s].y = rp[c + 1];
    }
  }

  float dk[KNN]; int ik[KNN];
#pragma unroll
  for (int t = 0; t < KNN; ++t) { dk[t] = -FLT_MAX; ik[t] = 0; }

  for (int m0 = 0; m0 < N_; m0 += 16) {
    // B fragments: columns m0..m0+15 (same striping, symmetric operand).
    v2f Bf[S];
    const float* cp = xb + (size_t)(m0 + lo) * CP;
#pragma unroll
    for (int s = 0; s < S; ++s) {
      int c = 4 * s + 2 * hi;
      Bf[s].x = cp[c];
      Bf[s].y = cp[c + 1];
    }
    if (m0 + 16 < N_)  // pull next column tile toward L0/L2 during top-k phase
      __builtin_prefetch((const void*)(cp + (size_t)16 * CP), 0, 1);

    v8f a0 = {}; v8f a1 = {};
#pragma unroll
    for (int s = 0; s < S; ++s) {
      a0 = __builtin_amdgcn_wmma_f32_16x16x4_f32(false, A[0][s], false, Bf[s],
                                                 (short)0, a0, false, false);
      a1 = __builtin_amdgcn_wmma_f32_16x16x4_f32(false, A[1][s], false, Bf[s],
                                                 (short)0, a1, false, false);
    }
    float xxc = xx[b * N_ + m0 + lo];
    // C/D layout: element j = row (j + 8*hi), column = lo. Spill to LDS so
    // each lane owns one full row for the selection phase.
#pragma unroll
    for (int j = 0; j < 8; ++j) {
      int M = j + 8 * hi;
      lds[(M)      * 16 + lo] = 2.f * a0[j] - xxc;
      lds[(16 + M) * 16 + lo] = 2.f * a1[j] - xxc;
    }
    __syncthreads();
#pragma unroll
    for (int j = 0; j < 16; ++j) {
      float d = lds[lane * 16 + j];
      int   m = m0 + j;
      if (d > dk[KNN - 1]) {            // sorted-descending insertion
        dk[KNN - 1] = d; ik[KNN - 1] = m;
#pragma unroll
        for (int t = KNN - 1; t > 0; --t) {
          if (dk[t] > dk[t - 1]) {
            float td = dk[t]; dk[t] = dk[t - 1]; dk[t - 1] = td;
            int   ti = ik[t]; ik[t] = ik[t - 1]; ik[t - 1] = ti;
          }
        }
      }
    }
    __syncthreads();
  }

  int n = r0 + lane;
  int* op = nbr + ((size_t)b * N_ + n) * KNN;
#pragma unroll
  for (int t = 0; t < KNN; ++t) op[t] = ik[t];
}

// ---------------- edge conv: gather + WMMA GEMM + LN + lrelu + max ---------
// 4 independent waves per block; each wave owns 16 flattened (n,k) rows.
// f[c] = c<C ? nbr[c]-ctr[c] : ctr[c-C].
template <int C, int CP, int TWOC, int O>
__global__ __launch_bounds__(128) void k_edgeconv(const float* __restrict__ xf,
                                                  const int* __restrict__ nbr,
                                                  const float* __restrict__ w,
                                                  const float* __restrict__ gg,
                                                  const float* __restrict__ bb,
                                                  float* __restrict__ y) {
  constexpr int S = (TWOC + 3) / 4;
  constexpr int T = O / 16;
  constexpr int TPB = (N_ * KNN) / 16;       // row tiles per batch
  int wid  = threadIdx.x >> 5;
  int tile = blockIdx.x * 4 + wid;           // total tiles divisible by 4
  int b    = tile / TPB;
  int r0   = (tile % TPB) * 16;
  int lane = threadIdx.x & 31;
  int lo   = lane & 15, hi = lane >> 4;

  // A fragment rows: lane supplies row lo, K elems {2*hi, 2*hi+1} per step.
  int r  = r0 + lo;
  int n  = r / KNN;
  int nb = nbr[(size_t)b * N_ * KNN + r];
  const float* ctr = xf + ((size_t)b * N_ + n)  * CP;
  const float* ngh = xf + ((size_t)b * N_ + nb) * CP;

  v2f A[S];
#pragma unroll
  for (int s = 0; s < S; ++s) {
    int c0 = 4 * s + 2 * hi;
    int c1 = c0 + 1;
    float v0 = (c0 < C) ? (ngh[c0] - ctr[c0]) : ((c0 < TWOC) ? ctr[c0 - C] : 0.f);
    float v1 = (c1 < C) ? (ngh[c1] - ctr[c1]) : ((c1 < TWOC) ? ctr[c1 - C] : 0.f);
    A[s].x = v0; A[s].y = v1;
  }

  v8f acc[T] = {};
#pragma unroll
  for (int t = 0; t < T; ++t) {
    int o = t * 16 + lo;
#pragma unroll
    for (int s = 0; s < S; ++s) {
      int c0 = 4 * s + 2 * hi;
      v2f Bf;
      Bf.x = (c0     < TWOC) ? w[o * TWOC + c0]     : 0.f;
      Bf.y = (c0 + 1 < TWOC) ? w[o * TWOC + c0 + 1] : 0.f;
      acc[t] = __builtin_amdgcn_wmma_f32_16x16x4_f32(false, A[s], false, Bf,
                                                     (short)0, acc[t], false, false);
    }
  }

  // Layer norm over the O output channels of each row. Row j's columns live
  // in element j across the 16 lanes of each half-wave -> width-16 xor-reduce.
  float mu[8], rs[8];
#pragma unroll
  for (int j = 0; j < 8; ++j) {
    float s = 0.f, ss = 0.f;
#pragma unroll
    for (int t = 0; t < T; ++t) { float v = acc[t][j]; s += v; ss += v * v; }
#pragma unroll
    for (int msk = 8; msk >= 1; msk >>= 1) {
      s  += __shfl_xor(s,  msk, 16);
      ss += __shfl_xor(ss, msk, 16);
    }
    float m   = s * (1.f / O);
    float var = ss * (1.f / O) - m * m;
    mu[j] = m;
    rs[j] = rsqrtf(var + EPS_);
  }

  float gv[T], bv[T];
#pragma unroll
  for (int t = 0; t < T; ++t) { int o = t * 16 + lo; gv[t] = gg[o]; bv[t] = bb[o]; }

  // Max over k: a lane's 8 rows span at most 2 distinct points, so keep a
  // running max and flush on point change -> <=2 atomics per lane per tile t
  // instead of 8. Max is associative: result & determinism unchanged.
#pragma unroll
  for (int t = 0; t < T; ++t) {
    int   cn = (r0 + 8 * hi) / KNN;       // point of row j=0
    float cm = -FLT_MAX;
#pragma unroll
    for (int j = 0; j < 8; ++j) {
      int rr = r0 + j + 8 * hi;
      int nn = rr / KNN;
      float v = (acc[t][j] - mu[j]) * rs[j] * gv[t] + bv[t];
      v = (v >= 0.f) ? v : SLOPE * v;
      if (nn != cn) {
        atomicMax(y + ((size_t)b * N_ + cn) * O + t * 16 + lo, cm);
        cn = nn; cm = v;
      } else {
        cm = fmaxf(cm, v);
      }
    }
    atomicMax(y + ((size_t)b * N_ + cn) * O + t * 16 + lo, cm);
  }
}

// ---------------- residual add ---------------------------------------------
__global__ void k_add(float* __restrict__ y, const float* __restrict__ x, int count) {
  int i = blockIdx.x * blockDim.x + threadIdx.x;
  if (i < count) y[i] += x[i];
}

// ---------------- skip 1x1 gemm add: y += x @ wa^T --------------------------
template <int CIN, int O>
__global__ void k_skip_add(float* __restrict__ y, const float* __restrict__ x,
                           const float* __restrict__ wa) {
  int i = blockIdx.x * blockDim.x + threadIdx.x;
  if (i >= B_ * N_ * O) return;
  int o = i % O;
  int p = i / O;
  const float* xp = x + (size_t)p * CIN;
  float s = 0.f;
#pragma unroll
  for (int c = 0; c < CIN; ++c) s += xp[c] * wa[o * CIN + c];
  y[i] += s;
}

// ---------------- concat + transpose to (B,128,N) --------------------------
__global__ void k_final(const float* __restrict__ x1, const float* __restrict__ x2,
                        const float* __restrict__ x3, const float* __restrict__ x4,
                        float* __restrict__ out) {
  int i = blockIdx.x * blockDim.x + threadIdx.x;
  if (i >= B_ * 128 * N_) return;
  int n = i % N_;
  int c = (i / N_) % 128;
  int b = i / (128 * N_);
  size_t p = (size_t)b * N_ + n;
  float v;
  if      (c < 16) v = x1[p * 16 + c];
  else if (c < 32) v = x2[p * 16 + (c - 16)];
  else if (c < 64) v = x3[p * 32 + (c - 32)];
  else             v = x4[p * 64 + (c - 64)];
  out[i] = v;
}

// ---------------------------------------------------------------------------
extern "C" void kernel_launch(void* const* d_in, const int* in_sizes, int n_in,
                              void* d_out, int out_size, void* d_ws, size_t ws_size,
                              hipStream_t stream) {
  const float* x   = (const float*)d_in[0];
  const float* w1  = (const float*)d_in[1];
  const float* g1  = (const float*)d_in[2];
  const float* b1  = (const float*)d_in[3];
  const float* w2  = (const float*)d_in[4];
  const float* g2  = (const float*)d_in[5];
  const float* b2  = (const float*)d_in[6];
  const float* w3  = (const float*)d_in[7];
  const float* g3  = (const float*)d_in[8];
  const float* b3  = (const float*)d_in[9];
  const float* w4  = (const float*)d_in[10];
  const float* g4  = (const float*)d_in[11];
  const float* b4  = (const float*)d_in[12];
  const float* wa1 = (const float*)d_in[13];
  const float* wa2 = (const float*)d_in[14];

  float* ws = (float*)d_ws;
  float* xt = ws;                               // B*N*4
  float* xx = xt + (size_t)B_ * N_ * 4;         // B*N
  int*   idx = (int*)(xx + (size_t)B_ * N_);    // B*N*KNN
  float* x1 = (float*)(idx + (size_t)B_ * N_ * KNN);  // B*N*16
  float* x2 = x1 + (size_t)B_ * N_ * 16;        // B*N*16
  float* x3 = x2 + (size_t)B_ * N_ * 16;        // B*N*32
  float* x4 = x3 + (size_t)B_ * N_ * 32;        // B*N*64

  const int BN = B_ * N_;
  const int knnBlocks  = B_ * (N_ / 32);
  const int convBlocks = B_ * (N_ * KNN / 16) / 4;   // 4 waves (tiles) / block

  k_transpose_pad<<<(BN + 255) / 256, 256, 0, stream>>>(x, xt);

  // ---- stage 1: 3 -> 16
  k_norms<<<(BN + 255) / 256, 256, 0, stream>>>(xt, 4, xx);
  k_knn<4><<<knnBlocks, 32, 0, stream>>>(xt, xx, idx);
  k_fill<<<(BN * 16 + 255) / 256, 256, 0, stream>>>(x1, BN * 16, -FLT_MAX);
  k_edgeconv<3, 4, 6, 16><<<convBlocks, 128, 0, stream>>>(xt, idx, w1, g1, b1, x1);

  // ---- stage 2: 16 -> 16 (+ residual)
  k_norms<<<(BN + 255) / 256, 256, 0, stream>>>(x1, 16, xx);
  k_knn<16><<<knnBlocks, 32, 0, stream>>>(x1, xx, idx);
  k_fill<<<(BN * 16 + 255) / 256, 256, 0, stream>>>(x2, BN * 16, -FLT_MAX);
  k_edgeconv<16, 16, 32, 16><<<convBlocks, 128, 0, stream>>>(x1, idx, w2, g2, b2, x2);
  k_add<<<(BN * 16 + 255) / 256, 256, 0, stream>>>(x2, x1, BN * 16);

  // ---- stage 3: 16 -> 32 (+ skip gemm wa1)
  k_norms<<<(BN + 255) / 256, 256, 0, stream>>>(x2, 16, xx);
  k_knn<16><<<knnBlocks, 32, 0, stream>>>(x2, xx, idx);
  k_fill<<<(BN * 32 + 255) / 256, 256, 0, stream>>>(x3, BN * 32, -FLT_MAX);
  k_edgeconv<16, 16, 32, 32><<<convBlocks, 128, 0, stream>>>(x2, idx, w3, g3, b3, x3);
  k_skip_add<16, 32><<<(BN * 32 + 255) / 256, 256, 0, stream>>>(x3, x2, wa1);

  // ---- stage 4: 32 -> 64 (+ skip gemm wa2)
  k_norms<<<(BN + 255) / 256, 256, 0, stream>>>(x3, 32, xx);
  k_knn<32><<<knnBlocks, 32, 0, stream>>>(x3, xx, idx);
  k_fill<<<(BN * 64 + 255) / 256, 256, 0, stream>>>(x4, BN * 64, -FLT_MAX);
  k_edgeconv<32, 32, 64, 64><<<convBlocks, 128, 0, stream>>>(x3, idx, w4, g4, b4, x4);
  k_skip_add<32, 64><<<(BN * 64 + 255) / 256, 256, 0, stream>>>(x4, x3, wa2);

  k_final<<<(B_ * 128 * N_ + 255) / 256, 256, 0, stream>>>(x1, x2, x3, x4, (float*)d_out);
}